// PUPHAWLoss_25709674234592
// MI455X (gfx1250) — compile-verified
//
#include <hip/hip_runtime.h>

typedef __attribute__((ext_vector_type(2))) float v2f;
typedef __attribute__((ext_vector_type(8))) float v8f;

#define FEAT 16
#define EPSF 1e-8f

__device__ __forceinline__ void atomicMaxNonneg(float* addr, float v) {
    // valid for non-negative floats: IEEE bit pattern is monotone as unsigned
    atomicMax((unsigned int*)addr, __float_as_uint(v));
}

// ---- K1: zero q[] and accumulators ----
__global__ void k_init(float* __restrict__ q, float* __restrict__ acc, int N) {
    int i = blockIdx.x * blockDim.x + threadIdx.x;
    if (i < N) q[i] = 0.0f;
    if (i < 8) acc[i] = 0.0f;
}

// ---- K2: per-edge d_e via f32 WMMA Gram diagonal, scatter-max into q[dst] ----
// One wave handles 16 edges. Dm = feats[src]-feats[dst] (16 edges x 16 feats).
// diag(Dm * Dm^T) = squared L2 norms, accumulated over four K=4 WMMA steps.
__global__ void k_edge(const float* __restrict__ feats, const float* __restrict__ pred,
                       const int* __restrict__ src, const int* __restrict__ dst,
                       float* __restrict__ q, int E) {
    const int lane = threadIdx.x & 31;
    const long long wid = ((long long)blockIdx.x * blockDim.x + threadIdx.x) >> 5;
    const long long eb = wid * 16;
    if (eb >= E) return;                 // wave-uniform: EXEC stays all-ones below

    const int m = lane & 15;             // edge-in-batch (row of Dm)
    int e = (int)eb + m;
    if (e >= E) e = E - 1;               // row-safe clamp; diag(m) depends only on row m
    const bool valid = ((long long)eb + m) < (long long)E;

    const int s = src[e], d = dst[e];
    const float* fs = feats + (size_t)s * FEAT;
    const float* fd = feats + (size_t)d * FEAT;
    const int khalf = (lane >> 4) << 1;  // lanes 0-15 -> K+{0,1}; lanes 16-31 -> K+{2,3}

    v8f c = {};
#pragma unroll
    for (int step = 0; step < 4; ++step) {
        const int k = step * 4 + khalf;
        float2 a2 = *(const float2*)(fs + k);
        float2 b2 = *(const float2*)(fd + k);
        v2f diff;
        diff[0] = a2.x - b2.x;
        diff[1] = a2.y - b2.y;
        // A (16x4) and B(=A^T, 4x16) fragments hold identical per-lane data here.
        c = __builtin_amdgcn_wmma_f32_16x16x4_f32(false, diff, false, diff,
                                                  (short)0, c, false, false);
    }

    // C layout: VGPR v, lanes 0-15 -> M=v, lanes 16-31 -> M=v+8; N = lane&15.
    // diag(m) sits at VGPR (m&7), lane (m<8 ? m : m+16).
    const int srcLane = (m < 8) ? m : (m + 16);
    float t0 = __shfl(c[0], srcLane, 32);
    float t1 = __shfl(c[1], srcLane, 32);
    float t2 = __shfl(c[2], srcLane, 32);
    float t3 = __shfl(c[3], srcLane, 32);
    float t4 = __shfl(c[4], srcLane, 32);
    float t5 = __shfl(c[5], srcLane, 32);
    float t6 = __shfl(c[6], srcLane, 32);
    float t7 = __shfl(c[7], srcLane, 32);
    const int v = m & 7;
    float x0 = (v & 1) ? t1 : t0;
    float x1 = (v & 1) ? t3 : t2;
    float x2 = (v & 1) ? t5 : t4;
    float x3 = (v & 1) ? t7 : t6;
    float y0 = (v & 2) ? x1 : x0;
    float y1 = (v & 2) ? x3 : x2;
    float norm2 = (v & 4) ? y1 : y0;

    float d_e = sqrtf(fmaxf(norm2, 0.0f)) + fabsf(pred[s] - pred[d]);
    if (lane < 16 && valid) atomicMaxNonneg(&q[d], d_e);
}

// ---- K3: sum(q) -> acc[0] ----
__global__ void k_sumq(const float* __restrict__ q, float* __restrict__ acc, int N) {
    __shared__ float sd[256];
    int i = blockIdx.x * blockDim.x + threadIdx.x;
    sd[threadIdx.x] = (i < N) ? q[i] : 0.0f;
    __syncthreads();
    for (int off = 128; off > 0; off >>= 1) {
        if (threadIdx.x < off) sd[threadIdx.x] += sd[threadIdx.x + off];
        __syncthreads();
    }
    if (threadIdx.x == 0) atomicAdd(&acc[0], sd[0]);
}

// ---- K4: w = 1 + q/(mean(q)+eps); zero m1 for hop-1 scatter ----
__global__ void k_weight(const float* __restrict__ q, const float* __restrict__ acc,
                         float* __restrict__ w, float* __restrict__ m1, int N) {
    int i = blockIdx.x * blockDim.x + threadIdx.x;
    if (i < N) {
        float meanq = acc[0] / (float)N;
        w[i] = 1.0f + q[i] / (meanq + EPSF);
        m1[i] = 0.0f;
    }
}

// ---- K5/K7: scatter-max mOut[dst[e]] = max(mOut[dst[e]], mIn[src[e]]) ----
__global__ void k_scatter_max(const float* __restrict__ mIn, const int* __restrict__ src,
                              const int* __restrict__ dst, float* __restrict__ mOut, int E) {
    int e = blockIdx.x * blockDim.x + threadIdx.x;
    if (e < E) atomicMaxNonneg(&mOut[dst[e]], mIn[src[e]]);
}

// ---- K6: w = max(w, 0.5*m1); zero q for reuse as m2 ----
__global__ void k_update1(float* __restrict__ w, const float* __restrict__ m1,
                          float* __restrict__ q, int N) {
    int i = blockIdx.x * blockDim.x + threadIdx.x;
    if (i < N) {
        w[i] = fmaxf(w[i], 0.5f * m1[i]);
        q[i] = 0.0f;
    }
}

// ---- K8: w = max(w, 0.25*m2) ----
__global__ void k_update2(float* __restrict__ w, const float* __restrict__ m2, int N) {
    int i = blockIdx.x * blockDim.x + threadIdx.x;
    if (i < N) w[i] = fmaxf(w[i], 0.25f * m2[i]);
}

// ---- K9: CSR SpMV residual + both loss partial sums -> acc[1], acc[2] ----
__global__ void k_final(const float* __restrict__ pred, const float* __restrict__ target,
                        const float* __restrict__ vals, const float* __restrict__ b,
                        const int* __restrict__ row_ptr, const int* __restrict__ col_ind,
                        const float* __restrict__ w, float* __restrict__ acc, int N) {
    __shared__ float sd[256];
    __shared__ float sp[256];
    int i = blockIdx.x * blockDim.x + threadIdx.x;
    float dloc = 0.0f, ploc = 0.0f;
    if (i < N) {
        float pd = pred[i] - target[i];
        dloc = pd * pd;
        float y = 0.0f;
        int k0 = row_ptr[i], k1 = row_ptr[i + 1];
        for (int k = k0; k < k1; ++k) y = fmaf(vals[k], pred[col_ind[k]], y);
        float r = y - b[i];
        ploc = w[i] * r * r;
    }
    sd[threadIdx.x] = dloc;
    sp[threadIdx.x] = ploc;
    __syncthreads();
    for (int off = 128; off > 0; off >>= 1) {
        if (threadIdx.x < off) {
            sd[threadIdx.x] += sd[threadIdx.x + off];
            sp[threadIdx.x] += sp[threadIdx.x + off];
        }
        __syncthreads();
    }
    if (threadIdx.x == 0) {
        atomicAdd(&acc[1], sd[0]);
        atomicAdd(&acc[2], sp[0]);
    }
}

// ---- K10: out = loss_data + loss_pde ----
__global__ void k_write(const float* __restrict__ acc, float* __restrict__ out, int N) {
    if (threadIdx.x == 0 && blockIdx.x == 0)
        out[0] = acc[1] / (float)N + acc[2] / (float)N;
}

extern "C" void kernel_launch(void* const* d_in, const int* in_sizes, int n_in,
                              void* d_out, int out_size, void* d_ws, size_t ws_size,
                              hipStream_t stream) {
    const float* pred    = (const float*)d_in[0];
    const float* target  = (const float*)d_in[1];
    const float* feats   = (const float*)d_in[2];
    const float* vals    = (const float*)d_in[3];
    const float* b       = (const float*)d_in[4];
    const int*   row_ptr = (const int*)d_in[5];
    const int*   col_ind = (const int*)d_in[6];
    // d_in[7] = row_idx (unused; row_ptr covers it), d_in[9] = epoch (unused)
    const int*   edge_index = (const int*)d_in[8];

    const int N = in_sizes[0];
    const int E = in_sizes[3];
    const int* src = edge_index;
    const int* dst = edge_index + E;

    float* q   = (float*)d_ws;   // also reused as m2 after K6
    float* w   = q + N;
    float* m1  = w + N;
    float* acc = m1 + N;         // 8 floats

    const int bn = (N + 255) / 256;
    const int be = (E + 255) / 256;
    const long long waves = ((long long)E + 15) / 16;
    const int bw = (int)((waves * 32 + 255) / 256);

    k_init       <<<bn, 256, 0, stream>>>(q, acc, N);
    k_edge       <<<bw, 256, 0, stream>>>(feats, pred, src, dst, q, E);
    k_sumq       <<<bn, 256, 0, stream>>>(q, acc, N);
    k_weight     <<<bn, 256, 0, stream>>>(q, acc, w, m1, N);
    k_scatter_max<<<be, 256, 0, stream>>>(w, src, dst, m1, E);   // hop 1
    k_update1    <<<bn, 256, 0, stream>>>(w, m1, q, N);
    k_scatter_max<<<be, 256, 0, stream>>>(m1, src, dst, q, E);   // hop 2
    k_update2    <<<bn, 256, 0, stream>>>(w, q, N);
    k_final      <<<bn, 256, 0, stream>>>(pred, target, vals, b, row_ptr, col_ind, w, acc, N);
    k_write      <<<1, 64, 0, stream>>>(acc, (float*)d_out, N);
}